// GKD_80942953660423
// MI455X (gfx1250) — compile-verified
//
#include <hip/hip_runtime.h>
#include <math.h>

#define B_ROWS 1024
#define L_CLS  32000
#define NT     256
#define SORT_N 32768
#define T_INV  0.25f   // 1/TEMPERATURE

typedef int v4i __attribute__((ext_vector_type(4)));

// ---- CDNA5 async global->LDS path (guarded; fallback = plain LDS store) ----
#if defined(__has_builtin)
# if __has_builtin(__builtin_amdgcn_global_load_async_to_lds_b128)
#  define USE_ASYNC_LDS 1
# endif
#endif
#ifndef USE_ASYNC_LDS
# define USE_ASYNC_LDS 0
#endif

__device__ __forceinline__ void async_row_b128(const float* g, float* l) {
#if USE_ASYNC_LDS
  __builtin_amdgcn_global_load_async_to_lds_b128(
      (__attribute__((address_space(1))) v4i*)g,
      (__attribute__((address_space(3))) v4i*)l, 0, 0);
#else
  *(float4*)l = *(const float4*)g;
#endif
}

__device__ __forceinline__ void wait_async_lds() {
#if USE_ASYNC_LDS
# if __has_builtin(__builtin_amdgcn_s_wait_asynccnt)
  __builtin_amdgcn_s_wait_asynccnt(0);
# else
  asm volatile("s_wait_asynccnt 0" ::: "memory");
# endif
#endif
}

// ---------------- block reduction helpers (256 threads) ----------------
__device__ __forceinline__ float blockSum256(float v, float* red, int tid) {
  red[tid] = v; __syncthreads();
  for (int off = 128; off > 0; off >>= 1) {
    if (tid < off) red[tid] += red[tid + off];
    __syncthreads();
  }
  float r = red[0]; __syncthreads();
  return r;
}

__device__ __forceinline__ float blockMax256(float v, float* red, int tid) {
  red[tid] = v; __syncthreads();
  for (int off = 128; off > 0; off >>= 1) {
    if (tid < off) red[tid] = fmaxf(red[tid], red[tid + off]);
    __syncthreads();
  }
  float r = red[0]; __syncthreads();
  return r;
}

// combine per-thread online logsumexp states (m, s)
__device__ __forceinline__ void blockLSE256(float m, float s, float* mred, float* sred,
                                            int tid, float* Mout, float* Sout) {
  mred[tid] = m; sred[tid] = s; __syncthreads();
  for (int off = 128; off > 0; off >>= 1) {
    if (tid < off) {
      float m1 = mred[tid], s1 = sred[tid];
      float m2 = mred[tid + off], s2 = sred[tid + off];
      float M = fmaxf(m1, m2);
      float S = 0.f;
      if (s1 > 0.f) S += s1 * expf(m1 - M);
      if (s2 > 0.f) S += s2 * expf(m2 - M);
      mred[tid] = M; sred[tid] = S;
    }
    __syncthreads();
  }
  *Mout = mred[0]; *Sout = sred[0]; __syncthreads();
}

__device__ __forceinline__ unsigned ordkey(float v) {
  unsigned u = __float_as_uint(v);
  return (u & 0x80000000u) ? ~u : (u | 0x80000000u);  // monotonic float->uint
}

// branch-free online logsumexp step
__device__ __forceinline__ void lse_step(float x, float& m, float& s) {
  float nm = fmaxf(m, x);
  s = s * expf(m - nm) + expf(x - nm);   // m==-inf: expf(-inf)=0, no NaN
  m = nm;
}

// ---------------- K1: row stats (T=1), copy-out, CE term ----------------
__global__ void k_rowstats(const float* __restrict__ ls, const int* __restrict__ tgt,
                           float* __restrict__ out, float* __restrict__ rowm,
                           float* __restrict__ rowz, float* __restrict__ cer) {
  __shared__ float mred[NT], sred[NT];
  int row = blockIdx.x, tid = threadIdx.x;
  const float* r = ls + (size_t)row * L_CLS;
  float* o = out + (size_t)row * L_CLS;
  float m = -INFINITY, s = 0.f;
  for (int i = tid * 4; i < L_CLS; i += NT * 4) {
    float4 x4 = *(const float4*)(r + i);
    *(float4*)(o + i) = x4;
    lse_step(x4.x, m, s);
    lse_step(x4.y, m, s);
    lse_step(x4.z, m, s);
    lse_step(x4.w, m, s);
  }
  float M, S;
  blockLSE256(m, s, mred, sred, tid, &M, &S);
  if (tid == 0) {
    rowm[row] = M; rowz[row] = S;
    int t = tgt[row];
    cer[row] = r[t] - M - logf(S);   // logp[target]
  }
}

// ---------------- K2: per-class softmax mass summed over batch ----------------
__global__ void k_colsum(const float* __restrict__ ls, const float* __restrict__ rowm,
                         const float* __restrict__ rowz, float* __restrict__ P) {
  __shared__ float ms[B_ROWS];
  __shared__ float zs[B_ROWS];
  int tid = threadIdx.x;
  for (int b = tid; b < B_ROWS; b += NT) { ms[b] = rowm[b]; zs[b] = rowz[b]; }
  __syncthreads();
  int c = blockIdx.x * NT + tid;
  const float* p = ls + c;
  float acc = 0.f;
  for (int b = 0; b < B_ROWS; ++b)
    acc += expf(p[(size_t)b * L_CLS] - ms[b]) / zs[b];
  P[c] = acc;
}

// ---------------- K3: descending sort + candidate-grid cumsum threshold ----------------
__global__ void k_sort_select(const float* __restrict__ P, float* __restrict__ buf,
                              int* __restrict__ hcm_out) {
  const int NTH = 1024;
  int tid = threadIdx.x;
  for (int i = tid; i < SORT_N; i += NTH) buf[i] = (i < L_CLS) ? P[i] : -1.0f;
  __threadfence_block(); __syncthreads();
  for (unsigned k = 2; k <= SORT_N; k <<= 1) {
    for (unsigned j = k >> 1; j > 0; j >>= 1) {
      for (unsigned i = tid; i < SORT_N; i += NTH) {
        unsigned ixj = i ^ j;
        if (ixj > i) {
          float a = buf[i], b = buf[ixj];
          bool up = ((i & k) == 0);           // descending overall
          if (up ? (a < b) : (a > b)) { buf[i] = b; buf[ixj] = a; }
        }
      }
      __threadfence_block(); __syncthreads();
    }
  }
  __shared__ double segd[320];
  if (tid < 320) {
    double ss = 0.0;
    const float* s = buf + tid * 100;
    for (int t = 0; t < 100; ++t) ss += (double)s[t];
    segd[tid] = ss;
  }
  __syncthreads();
  if (tid == 0) {
    const double TH = (double)(0.93f * 1024.0f);   // RATE * B
    double run = 0.0; int hcm = 1; bool found = false;
    for (int j = 0; j < 320; ++j) {                // cand = 100*j + 1
      double cs = run + (double)buf[j * 100];      // cumsum through element 100j
      if (!found && cs >= TH) { hcm = j * 100 + 1; found = true; }
      run += segd[j];
    }
    *hcm_out = hcm;
  }
}

// ---------------- K4: per-row mask (radix-select) + all KL terms ----------------
// LDS: student row + teacher row staged via CDNA5 async-to-LDS, all passes LDS-resident.
__global__ void k_gkd_row(const float* __restrict__ ls, const float* __restrict__ lt,
                          const int* __restrict__ tgt, const int* __restrict__ hcm_p,
                          float* __restrict__ bin_row, float* __restrict__ hard_row) {
  extern __shared__ char smem[];
  float*    xs   = (float*)smem;                 // 32000 : student row
  float*    ys   = xs + L_CLS;                   // 32000 : teacher row
  unsigned* hist = (unsigned*)(ys + L_CLS);      // 256
  float*    redA = (float*)(hist + 256);         // 256
  float*    redB = redA + 256;                   // 256
  int*      eq   = (int*)(redB + 256);           // 64
  __shared__ int scI[4];                         // sel, kk, eq-count, cutoff

  int row = blockIdx.x, tid = threadIdx.x;
  int t   = tgt[row];
  int hcm = *hcm_p;
  const float* rs = ls + (size_t)row * L_CLS;
  const float* rt = lt + (size_t)row * L_CLS;

  // ---- P0: DMA both rows into LDS (overlapped), single wait ----
  for (int i = tid * 4; i < L_CLS; i += NT * 4) {
    async_row_b128(rs + i, xs + i);
    async_row_b128(rt + i, ys + i);
  }
  wait_async_lds();
  __syncthreads();

  // full-row online LSE of ls/T (branch-free)
  float m = -INFINITY, s = 0.f;
  for (int c = tid; c < L_CLS; c += NT)
    lse_step(xs[c] * T_INV, m, s);
  float m4s, Zs4;
  blockLSE256(m, s, redA, redB, tid, &m4s, &Zs4);

  // ---- radix select: hcm-th largest key of class_select ----
  unsigned prefix = 0; int kk = hcm;
  for (int pass = 0; pass < 4; ++pass) {
    int shift = 24 - 8 * pass;
    if (tid < 256) hist[tid] = 0;
    __syncthreads();
    for (int c = tid; c < L_CLS; c += NT) {
      float v = (c == t) ? 999999.0f : xs[c];
      unsigned u = ordkey(v);
      if (pass == 0 || (u >> (shift + 8)) == prefix)
        atomicAdd(&hist[(u >> shift) & 255u], 1u);
    }
    __syncthreads();
    if (tid == 0) {
      unsigned acc = 0;
      for (int d = 255; d >= 0; --d) {
        acc += hist[d];
        if ((int)acc >= kk) { scI[0] = d; scI[1] = kk - (int)(acc - hist[d]); break; }
      }
    }
    __syncthreads();
    prefix = (prefix << 8) | (unsigned)scI[0];
    kk = scI[1];
    __syncthreads();
  }
  unsigned u_k = prefix;
  int need = kk;

  // ---- tie-break by index (stable argsort semantics) ----
  if (tid == 0) scI[2] = 0;
  __syncthreads();
  for (int c = tid; c < L_CLS; c += NT) {
    float v = (c == t) ? 999999.0f : xs[c];
    if (ordkey(v) == u_k) { int p = atomicAdd(&scI[2], 1); if (p < 64) eq[p] = c; }
  }
  __syncthreads();
  if (tid == 0) {
    int n = scI[2]; if (n > 64) n = 64;
    for (int a = 1; a < n; ++a) {               // insertion sort (tiny)
      int v = eq[a]; int b = a - 1;
      while (b >= 0 && eq[b] > v) { eq[b + 1] = eq[b]; --b; }
      eq[b + 1] = v;
    }
    scI[3] = (need <= n) ? eq[need - 1] : L_CLS;
  }
  __syncthreads();
  int cutoff = scI[3];

  auto in_hard = [&](int c, float xv) -> bool {
    float v = (c == t) ? 999999.0f : xv;
    unsigned u = ordkey(v);
    return (u > u_k) || (u == u_k && c <= cutoff);
  };

  // ---- masked maxes (student hard, teacher hard) + teacher full LSE ----
  float mm = -INFINITY, mt = -INFINITY, st = 0.f, mth = -INFINITY;
  for (int c = tid; c < L_CLS; c += NT) {
    float xv = xs[c];
    float y  = ys[c] * T_INV;
    bool  h  = in_hard(c, xv);
    lse_step(y, mt, st);
    mm  = fmaxf(mm,  h ? xv * T_INV : -INFINITY);
    mth = fmaxf(mth, h ? y          : -INFINITY);
  }
  float m_sh = blockMax256(mm, redA, tid);
  float m4t, Zt4;
  blockLSE256(mt, st, redA, redB, tid, &m4t, &Zt4);
  float m_th = blockMax256(mth, redA, tid);

  // ---- group sums + hard-group KL pieces (branch-free, LDS-resident) ----
  float sh = 0.f, sn = 0.f, zsh = 0.f;
  float shT = 0.f, snT = 0.f, zth = 0.f, Aacc = 0.f;
  for (int c = tid; c < L_CLS; c += NT) {
    float xv = xs[c];
    float x4 = xv * T_INV;
    float y  = ys[c] * T_INV;
    bool  h  = in_hard(c, xv);
    float e0 = expf(x4 - m4s);        // student full-softmax numerator
    float e1 = expf(x4 - m_sh);       // student hard-softmax numerator
    float f0 = expf(y - m4t);         // teacher full-softmax numerator
    float f1 = expf(y - m_th);        // teacher hard-softmax numerator
    sh  += h ? e0 : 0.f;
    sn  += h ? 0.f : e0;
    zsh += h ? e1 : 0.f;
    shT += h ? f0 : 0.f;
    snT += h ? 0.f : f0;
    zth += h ? f1 : 0.f;
    Aacc += h ? f1 * (y - m_th - x4) : 0.f;
  }
  float Sh  = blockSum256(sh,  redA, tid);
  float Sn  = blockSum256(sn,  redA, tid);
  float Zsh = blockSum256(zsh, redA, tid);
  float ShT = blockSum256(shT, redA, tid);
  float SnT = blockSum256(snT, redA, tid);
  float Zth = blockSum256(zth, redA, tid);
  float A   = blockSum256(Aacc, redA, tid);

  if (tid == 0) {
    float ps_h = Sh / Zs4, ps_n = Sn / Zs4;
    float pt_h = ShT / Zt4, pt_n = SnT / Zt4;
    float bl = 0.f;
    if (pt_h > 0.f) bl += pt_h * (logf(pt_h) - logf(ps_h));
    if (pt_n > 0.f) bl += pt_n * (logf(pt_n) - logf(ps_n));
    bin_row[row] = bl;
    // KL(pth || exp(lsh)) over hard set = A/Zth + (m_sh + ln Zsh - ln Zth)
    float hl = 0.f;
    if (Zth > 0.f) hl = A / Zth + (m_sh + logf(Zsh) - logf(Zth));
    hard_row[row] = hl;
  }
}

// ---------------- K5: finalize scalars ----------------
__global__ void k_final(const float* __restrict__ cer, const float* __restrict__ bin_row,
                        const float* __restrict__ hard_row, const int* __restrict__ ep,
                        float* __restrict__ out_sc) {
  __shared__ float red[NT];
  int tid = threadIdx.x;
  float a = 0.f, b = 0.f, c = 0.f;
  for (int i = tid; i < B_ROWS; i += NT) { a += cer[i]; b += bin_row[i]; c += hard_row[i]; }
  float sa = blockSum256(a, red, tid);
  float sb = blockSum256(b, red, tid);
  float sc = blockSum256(c, red, tid);
  if (tid == 0) {
    float loss_ce = -(sa / (float)B_ROWS);                 // CE_W = 1
    float scale = 16.0f / (float)B_ROWS;                   // T^2 / B
    float binary_loss = sb * scale;
    float hard_loss   = sc * scale;
    float warm = fminf((float)ep[0] / 10.0f, 1.0f);        // WARMUP = 10
    out_sc[0] = loss_ce;
    out_sc[1] = warm * (1.0f * binary_loss + 4.0f * hard_loss);  // ALPHA, BETA (LAMDA=0)
  }
}

extern "C" void kernel_launch(void* const* d_in, const int* in_sizes, int n_in,
                              void* d_out, int out_size, void* d_ws, size_t ws_size,
                              hipStream_t stream) {
  const float* ls  = (const float*)d_in[0];
  const float* lt  = (const float*)d_in[1];
  const int*   tgt = (const int*)d_in[2];
  const int*   ep  = (const int*)d_in[3];
  float* out = (float*)d_out;

  float* ws      = (float*)d_ws;
  float* rowm    = ws;                 // 1024
  float* rowz    = ws + 1024;          // 1024
  float* cer     = ws + 2048;          // 1024
  float* binr    = ws + 3072;          // 1024
  float* hardr   = ws + 4096;          // 1024
  float* P       = ws + 5120;          // 32000
  float* sortbuf = P + L_CLS;          // 32768
  int*   hcm     = (int*)(sortbuf + SORT_N);

  k_rowstats<<<B_ROWS, NT, 0, stream>>>(ls, tgt, out, rowm, rowz, cer);
  k_colsum<<<L_CLS / NT, NT, 0, stream>>>(ls, rowm, rowz, P);
  k_sort_select<<<1, 1024, 0, stream>>>(P, sortbuf, hcm);
  size_t smem = (size_t)(2 * L_CLS + 256 + 256 + 256 + 64) * sizeof(float);
  k_gkd_row<<<B_ROWS, NT, smem, stream>>>(ls, lt, tgt, hcm, binr, hardr);
  k_final<<<1, NT, 0, stream>>>(cer, binr, hardr, ep, out + (size_t)B_ROWS * L_CLS);
}